// AttentionFilter_21698174780222
// MI455X (gfx1250) — compile-verified
//
#include <hip/hip_runtime.h>
#include <hip/hip_bf16.h>

typedef __attribute__((ext_vector_type(2))) float v2f;
typedef __attribute__((ext_vector_type(8))) float v8f;
typedef __attribute__((ext_vector_type(4))) int   v4i;

#define NROWS 4096
#define INF 512
#define OUTF 64
#define NEG_SLOPE 0.01f
#define MASK_VAL -9.0e15f

#if __has_builtin(__builtin_amdgcn_global_load_async_to_lds_b128) && \
    __has_builtin(__builtin_amdgcn_s_wait_asynccnt)
#define HAVE_ASYNC_LDS 1
#else
#define HAVE_ASYNC_LDS 0
#endif

// ---------------------------------------------------------------------------
// Kernel A: build zero-padded B panel bpad[16][512] (row-major, row n = column
// n of the [512,16] B matrix). Row 0 = W @ a[:64], row 1 = W @ a[64:], rows
// 2..15 = 0.  This lets kernel B load B fragments unconditionally (no EXEC
// juggling inside the WMMA K-loop).
// ---------------------------------------------------------------------------
__global__ void gat_wa_kernel(const float* __restrict__ W,   // [512,64]
                              const float* __restrict__ a,   // [128]
                              float* __restrict__ bpad)      // [16*512]
{
    int idx = blockIdx.x * blockDim.x + threadIdx.x;  // 0 .. 8191
    int n = idx >> 9;
    int k = idx & (INF - 1);
    float s = 0.f;
    if (n < 2) {
        const float* w  = W + k * OUTF;
        const float* av = a + n * OUTF;
#pragma unroll 8
        for (int j = 0; j < OUTF; ++j) s = fmaf(w[j], av[j], s);
    }
    bpad[idx] = s;
}

// ---------------------------------------------------------------------------
// Kernel B: F = X @ B via V_WMMA_F32_16X16X4_F32, one wave per 16-row tile.
// A-tile (16x4 f32): lanes 0-15 -> M=lane, K=k..k+1; lanes 16-31 -> M=lane-16,
// K=k+2..k+3.  B-tile (4x16): symmetric with lane=N.  C/D (16x16 f32, 8
// VGPRs): VGPR r: lanes 0-15 -> (M=r, N=lane); lanes 16-31 -> (M=r+8, N-16).
// Unconditional b64 A/B fragment loads; unroll 8 so loads batch ahead of a
// run of back-to-back WMMAs.
// ---------------------------------------------------------------------------
__global__ void gat_gemv_wmma_kernel(const float* __restrict__ X,     // [4096,512]
                                     const float* __restrict__ bpad,  // [16*512]
                                     float* __restrict__ f)           // [8192]: f1|f2
{
    const int lane    = threadIdx.x;        // 0..31
    const int mn      = lane & 15;          // M for A-side, N for B-side
    const int khalf   = lane >> 4;          // 0 -> K pair {0,1}; 1 -> {2,3}
    const int rowBase = blockIdx.x * 16;

    const float2* __restrict__ x2 =
        reinterpret_cast<const float2*>(X + (size_t)(rowBase + mn) * INF);
    const float2* __restrict__ b2 =
        reinterpret_cast<const float2*>(bpad + mn * INF);

    v8f acc = {};
#pragma unroll 8
    for (int k2 = 0; k2 < INF / 2; k2 += 2) {   // K chunk of 4 per iteration
        float2 xa = x2[k2 + khalf];
        float2 xb = b2[k2 + khalf];
        v2f av; av.x = xa.x; av.y = xa.y;
        v2f bv; bv.x = xb.x; bv.y = xb.y;
        acc = __builtin_amdgcn_wmma_f32_16x16x4_f32(
            /*neg_a=*/false, av, /*neg_b=*/false, bv,
            /*c_mod=*/(short)0, acc, /*reuse_a=*/false, /*reuse_b=*/false);
    }

    // Columns 0/1 of the D tile are f1/f2 for these 16 rows.
    if (mn < 2) {
        float* dst = f + mn * NROWS + rowBase + khalf * 8;
#pragma unroll
        for (int r = 0; r < 8; ++r) dst[r] = acc[r];
    }
}

// ---------------------------------------------------------------------------
// Kernel C: per-row fused leaky-relu -> adjacency mask -> softmax.
// One block (256 threads = 8 waves) per row.  The adj row (16 KB) is staged
// into LDS with async global->LDS b128 loads (ASYNCcnt) when available, so
// each wave keeps its full HBM stream in flight with no VGPR cost; scores are
// cached in LDS so adj is read from HBM exactly once and out written once.
// ---------------------------------------------------------------------------
__device__ __forceinline__ float score1(float e, int adjv) {
    e = (e >= 0.f) ? e : NEG_SLOPE * e;
    return (adjv > 0) ? e : MASK_VAL;
}

__device__ __forceinline__ float waveMax(float v) {
#pragma unroll
    for (int o = 16; o > 0; o >>= 1) v = fmaxf(v, __shfl_xor(v, o, 32));
    return v;
}
__device__ __forceinline__ float waveSum(float v) {
#pragma unroll
    for (int o = 16; o > 0; o >>= 1) v += __shfl_xor(v, o, 32);
    return v;
}

__global__ void gat_softmax_kernel(const int* __restrict__ adj,  // [4096,4096]
                                   const float* __restrict__ f,  // f1 | f2
                                   float* __restrict__ out)      // [4096,4096]
{
    __shared__ int   s_adj[NROWS];         // 16 KB
    __shared__ float s_scores[NROWS];      // 16 KB
    __shared__ float s_red[8];

    const int row = blockIdx.x;
    const int tid = threadIdx.x;
    const int wid = tid >> 5;
    const int lid = tid & 31;

    const int4* __restrict__ arow4 =
        reinterpret_cast<const int4*>(adj + (size_t)row * NROWS);
    int4* s_adj4 = reinterpret_cast<int4*>(s_adj);

#if HAVE_ASYNC_LDS
    // Stage this wave's share of the adj row into LDS asynchronously.
#pragma unroll
    for (int i = tid; i < NROWS / 4; i += 256) {
        __builtin_amdgcn_global_load_async_to_lds_b128(
            (v4i*)(arow4 + i),
            (v4i*)(s_adj4 + i),
            /*offset=*/0, /*cpol=*/0);
    }
#endif

    const float f1i = f[row];
    const float4* __restrict__ f2v = reinterpret_cast<const float4*>(f + NROWS);
    float4* __restrict__ out4 =
        reinterpret_cast<float4*>(out + (size_t)row * NROWS);
    float4* s4 = reinterpret_cast<float4*>(s_scores);

#if HAVE_ASYNC_LDS
    __builtin_amdgcn_s_wait_asynccnt(0);   // this wave's LDS data is resident
#endif

    // Pass 1: scores -> LDS, track running max.
    float m = -3.0e38f;
#pragma unroll
    for (int i = tid; i < NROWS / 4; i += 256) {
#if HAVE_ASYNC_LDS
        int4 av = s_adj4[i];
#else
        int4 av = arow4[i];
#endif
        float4 fv = f2v[i];
        float4 s;
        s.x = score1(f1i + fv.x, av.x);
        s.y = score1(f1i + fv.y, av.y);
        s.z = score1(f1i + fv.z, av.z);
        s.w = score1(f1i + fv.w, av.w);
        s4[i] = s;
        m = fmaxf(m, fmaxf(fmaxf(s.x, s.y), fmaxf(s.z, s.w)));
    }
    m = waveMax(m);
    if (lid == 0) s_red[wid] = m;
    __syncthreads();
    {
        float v = (lid < 8) ? s_red[lid] : -3.0e38f;
        v = waveMax(v);
        if (tid == 0) s_red[0] = v;
    }
    __syncthreads();
    const float rowMax = s_red[0];
    __syncthreads();

    // Pass 2: exponentiate in LDS, accumulate sum.
    float sum = 0.f;
#pragma unroll
    for (int i = tid; i < NROWS / 4; i += 256) {
        float4 s = s4[i];
        float4 ex;
        ex.x = __expf(s.x - rowMax);
        ex.y = __expf(s.y - rowMax);
        ex.z = __expf(s.z - rowMax);
        ex.w = __expf(s.w - rowMax);
        s4[i] = ex;
        sum += (ex.x + ex.y) + (ex.z + ex.w);
    }
    sum = waveSum(sum);
    if (lid == 0) s_red[wid] = sum;
    __syncthreads();
    {
        float v = (lid < 8) ? s_red[lid] : 0.f;
        v = waveSum(v);
        if (tid == 0) s_red[0] = v;
    }
    __syncthreads();
    const float inv = 1.0f / s_red[0];

    // Pass 3: normalized write-out (b128 stores).
#pragma unroll
    for (int i = tid; i < NROWS / 4; i += 256) {
        float4 ex = s4[i];
        float4 o;
        o.x = ex.x * inv; o.y = ex.y * inv; o.z = ex.z * inv; o.w = ex.w * inv;
        out4[i] = o;
    }
}

// ---------------------------------------------------------------------------
extern "C" void kernel_launch(void* const* d_in, const int* in_sizes, int n_in,
                              void* d_out, int out_size, void* d_ws, size_t ws_size,
                              hipStream_t stream) {
    const float* X   = (const float*)d_in[0];   // [4096,512]
    const int*   adj = (const int*)d_in[1];     // [4096,4096]
    const float* W   = (const float*)d_in[2];   // [512,64]
    const float* a   = (const float*)d_in[3];   // [128,1]
    float* out = (float*)d_out;                 // [4096,4096]

    float* bpad = (float*)d_ws;                 // 16*512 floats (32 KB)
    float* f    = bpad + 16 * INF;              // 8192 floats (f1 | f2)

    gat_wa_kernel<<<(16 * INF) / 256, 256, 0, stream>>>(W, a, bpad);
    gat_gemv_wmma_kernel<<<NROWS / 16, 32, 0, stream>>>(X, bpad, f);
    gat_softmax_kernel<<<NROWS, 256, 0, stream>>>(adj, f, out);
}